// FastAGaLiTeLayer_25993142076004
// MI455X (gfx1250) — compile-verified
//
#include <hip/hip_runtime.h>
#include <math.h>

// ---------------------------------------------------------------------------
// FastAGaLiTe layer for gfx1250 (MI455X), fp32 end-to-end (matches reference).
//   K1: all_proj = inputs @ Wf^T + bf          (WMMA f32 16x16x4 GEMM,
//                                               async global->LDS staging)
//   K2: per-(b,h) recurrent scan over T=128    (register-resident state)
//   K3: out = attn_out @ Wp^T + bp             (same WMMA GEMM)
// ---------------------------------------------------------------------------

typedef __attribute__((ext_vector_type(2))) float v2f;
typedef __attribute__((ext_vector_type(8))) float v8f;
typedef __attribute__((ext_vector_type(4))) int   v4i;

#define T_LEN 128
#define BATCH 32
#define DMODEL 1024
#define NHEAD 8
#define DHEAD 64
#define ETA 4
#define NR 4
#define FDIM 256           // DHEAD * ETA
#define PROJ 2656          // H*DH*5 + H*ETA*3
#define KQV_DIM 2560       // H*DH*5
#define EPS 1e-6f
#define PI_F 3.14159265358979323846f

#define TILE_M 128
#define TILE_N 32
#define KCH 32
#define LDSP 36            // padded LDS row pitch in floats: 144B = 16B-aligned,
                           // and 36*r mod 64 distinct for r=0..15 (no bank conflicts)

// ---- gfx1250 async global->LDS staging (ASYNCcnt path), with fallback ----
#if __has_builtin(__builtin_amdgcn_global_load_async_to_lds_b128)
#define HAVE_ASYNC_LDS 1
#else
#define HAVE_ASYNC_LDS 0
#endif

typedef __attribute__((address_space(1))) v4i g_v4i;
typedef __attribute__((address_space(3))) v4i l_v4i;

__device__ __forceinline__ void copy16_to_lds(const float* g, float* l) {
#if HAVE_ASYNC_LDS
  __builtin_amdgcn_global_load_async_to_lds_b128(
      (g_v4i*)g, (l_v4i*)l, /*offset=*/0, /*cpol=*/0);
#else
  *(float4*)l = *(const float4*)g;
#endif
}

__device__ __forceinline__ void wait_async_zero() {
#if HAVE_ASYNC_LDS
#if __has_builtin(__builtin_amdgcn_s_wait_asynccnt)
  __builtin_amdgcn_s_wait_asynccnt(0);
#else
  asm volatile("s_wait_asynccnt 0x0" ::: "memory");
#endif
#endif
}

// ---------------------------------------------------------------------------
// K1/K3: C[M,N] = A[M,K] @ B[N,K]^T + bias[N]   (A,B row-major, K contiguous)
// 256 threads = 8 waves; block tile 128(M) x 32(N); wave w owns rows
// [w*16, w*16+16) x all 32 cols: two 16x16 WMMA accumulators sharing one A frag.
// Both A and B tiles are staged as straight row copies (async b128 -> LDS);
// the B operand is read as Bs[n][k] which is exactly Wf[n][k] = (B^T)[k][n].
// ---------------------------------------------------------------------------
__global__ __launch_bounds__(256)
void wmma_gemm_nt(const float* __restrict__ A, const float* __restrict__ Bw,
                  const float* __restrict__ bias, float* __restrict__ C,
                  int M, int N, int K) {
  __shared__ float As[TILE_M][LDSP];   // 128 x 36 floats (18 KB)
  __shared__ float Bs[TILE_N][LDSP];   //  32 x 36 floats (4.5 KB)

  const int tid  = threadIdx.x;
  const int wave = tid >> 5;     // 0..7 -> 16-row strip
  const int lane = tid & 31;
  const int row  = lane & 15;
  const int sel  = lane >> 4;    // 0/1: K-half selector (A,B), M-half selector (C)

  const int n0 = blockIdx.x * TILE_N;
  const int m0 = blockIdx.y * TILE_M;

  const int bn  = tid >> 3;      // 0..31  B staging row
  const int bkg = tid & 7;       // 16B group within K-chunk

  v8f acc0 = {};                 // cols n0 + [0,16)
  v8f acc1 = {};                 // cols n0 + [16,32)

  for (int k0 = 0; k0 < K; k0 += KCH) {
    // ---- stage B tile: Bs[n][k] = Bw[n0+n][k0+k]  (1 x b128 per thread) ----
    copy16_to_lds(Bw + (size_t)(n0 + bn) * K + k0 + bkg * 4, &Bs[bn][bkg * 4]);
    // ---- stage A tile: As[m][k] = A[m0+m][k0+k]   (4 x b128 per thread) ----
#pragma unroll
    for (int i = 0; i < 4; ++i) {
      const int idx = tid + 256 * i;
      const int am  = idx >> 3;        // 0..127
      const int akg = idx & 7;
      copy16_to_lds(A + (size_t)(m0 + am) * K + k0 + akg * 4, &As[am][akg * 4]);
    }
    // prefetch next K-chunk into cache (gfx1250 global_prefetch_b8 path)
    if (k0 + KCH < K) {
      __builtin_prefetch(A + (size_t)(m0 + (tid & 127)) * K + k0 + KCH, 0, 3);
      __builtin_prefetch(Bw + (size_t)(n0 + (tid & 31)) * K + k0 + KCH, 0, 3);
    }
    wait_async_zero();
    __syncthreads();

    // ---- compute: 16 x V_WMMA_F32_16X16X4_F32 per chunk per wave ----
#pragma unroll
    for (int k4 = 0; k4 < KCH; k4 += 4) {
      v2f a, b0, b1;
      // A 16x4: lanes 0-15 -> K = {k4,k4+1}; lanes 16-31 -> K = {k4+2,k4+3}
      a.x  = As[wave * 16 + row][k4 + sel * 2 + 0];
      a.y  = As[wave * 16 + row][k4 + sel * 2 + 1];
      // B 4x16: lane = N index; VGPR pair holds the two K values of its half
      b0.x = Bs[row][k4 + sel * 2 + 0];
      b0.y = Bs[row][k4 + sel * 2 + 1];
      b1.x = Bs[16 + row][k4 + sel * 2 + 0];
      b1.y = Bs[16 + row][k4 + sel * 2 + 1];
      acc0 = __builtin_amdgcn_wmma_f32_16x16x4_f32(false, a, false, b0,
                                                   (short)0, acc0, false, false);
      acc1 = __builtin_amdgcn_wmma_f32_16x16x4_f32(false, a, false, b1,
                                                   (short)0, acc1, false, false);
    }
    __syncthreads();
  }

  // ---- store: C/D layout: VGPR i holds (M = i + sel*8, N = lane&15) ----
  const int cn0 = n0 + row;
  const int cn1 = n0 + 16 + row;
  const float bv0 = bias[cn0];
  const float bv1 = bias[cn1];
#pragma unroll
  for (int i = 0; i < 8; ++i) {
    const int cm = m0 + wave * 16 + sel * 8 + i;
    C[(size_t)cm * N + cn0] = acc0[i] + bv0;
    C[(size_t)cm * N + cn1] = acc1[i] + bv1;
  }
}

// ---------------------------------------------------------------------------
// K2: recurrent scan.  grid = B (32 blocks), block = 256 threads = 8 waves.
// Wave h owns head h of batch b.  Per-lane state strips (f = j*32+lane):
//   Kst[R][8] (R x FDIM), Vst[R][2] (R x DHEAD), Sst[8] (FDIM).
// ---------------------------------------------------------------------------
__device__ __forceinline__ float sigm(float x) { return 1.0f / (1.0f + expf(-x)); }

__global__ __launch_bounds__(256)
void agalite_scan(const float* __restrict__ proj,    // (T,B,PROJ)
                  const int*   __restrict__ term,    // (T,B)
                  const float* __restrict__ k_prev,  // (B,R,H,FD)
                  const float* __restrict__ v_prev,  // (B,R,H,DH)
                  const float* __restrict__ s_prev,  // (B,H,FD)
                  const float* __restrict__ tick_in, // (B)
                  float* __restrict__ attn,          // (T,B,H*DH)
                  float* __restrict__ fk_out,        // (B,R,H,FD)
                  float* __restrict__ fv_out,        // (B,R,H,DH)
                  float* __restrict__ fs_out,        // (B,H,FD)
                  float* __restrict__ tick_out) {    // (B)
  const int b    = blockIdx.x;
  const int h    = threadIdx.x >> 5;
  const int lane = threadIdx.x & 31;

  __shared__ float sk[NHEAD][DHEAD];   // relu(keys)
  __shared__ float sq[NHEAD][DHEAD];   // relu(queries)
  __shared__ float sg[NHEAD][DHEAD];   // sigmoid(gammas)
  __shared__ float sc[NHEAD][NR];      // per-head partial scores

  float Kst[NR][8], Vst[NR][2], Sst[8];

  // ---- load carried state ----
#pragma unroll
  for (int r = 0; r < NR; ++r) {
#pragma unroll
    for (int j = 0; j < 8; ++j) {
      const int f = j * 32 + lane;
      Kst[r][j] = k_prev[((size_t)(b * NR + r) * NHEAD + h) * FDIM + f];
    }
#pragma unroll
    for (int j = 0; j < 2; ++j) {
      const int d = j * 32 + lane;
      Vst[r][j] = v_prev[((size_t)(b * NR + r) * NHEAD + h) * DHEAD + d];
    }
  }
#pragma unroll
  for (int j = 0; j < 8; ++j) {
    const int f = j * 32 + lane;
    Sst[j] = s_prev[((size_t)b * NHEAD + h) * FDIM + f];
  }

  const float tk = tick_in[b];
  const float omega[NR] = {-PI_F, -PI_F / 3.0f, PI_F / 3.0f, PI_F};

  for (int t = 0; t < T_LEN; ++t) {
    const float* pb = proj + ((size_t)t * BATCH + b) * PROJ;

    // ---- stage activated per-head vectors into LDS ----
    float vv[2], bb[2];
#pragma unroll
    for (int j = 0; j < 2; ++j) {
      const int d = j * 32 + lane;
      const float kk = pb[(h * 5 + 0) * DHEAD + d];
      const float qq = pb[(h * 5 + 1) * DHEAD + d];
      vv[j]          = pb[(h * 5 + 2) * DHEAD + d];
      bb[j]     = sigm(pb[(h * 5 + 3) * DHEAD + d]);
      const float gg = pb[(h * 5 + 4) * DHEAD + d];
      sk[h][d] = fmaxf(kk, 0.0f);
      sq[h][d] = fmaxf(qq, 0.0f);
      sg[h][d] = sigm(gg);
    }
    float rp1[ETA], rp2[ETA], sp3[ETA];
#pragma unroll
    for (int e = 0; e < ETA; ++e) {
      rp1[e] = fmaxf(pb[KQV_DIM + (h * 3 + 0) * ETA + e], 0.0f);
      rp2[e] = fmaxf(pb[KQV_DIM + (h * 3 + 1) * ETA + e], 0.0f);
      sp3[e] = sigm (pb[KQV_DIM + (h * 3 + 2) * ETA + e]);
    }
    const float mask = 1.0f - (float)term[t * BATCH + b];
    const float tt = (float)(t + 1) + tk;
    float cs[NR];
#pragma unroll
    for (int r = 0; r < NR; ++r) cs[r] = cosf(tt * omega[r]);

    __syncthreads();

    // ---- state updates + per-lane partial reductions ----
    float scoreL[NR] = {0.0f, 0.0f, 0.0f, 0.0f};
    float normL = 0.0f;
#pragma unroll
    for (int j = 0; j < 8; ++j) {
      const int f = j * 32 + lane;
      const int d = f >> 2;
      const int e = f & 3;
      const float psi = sk[h][d] * rp1[e];
      const float gf  = sg[h][d] * sp3[e];
      const float kg  = psi * gf;                 // keys_gated == s
      const float dg  = (1.0f - gf) * mask;       // discount_gamma
      const float phi = sq[h][d] * rp2[e];        // phi_q
      Sst[j] = dg * Sst[j] + kg;
      normL += Sst[j] * phi;
#pragma unroll
      for (int r = 0; r < NR; ++r) {
        Kst[r][j] = dg * Kst[r][j] + kg * cs[r];
        scoreL[r] += Kst[r][j] * phi;
      }
    }
#pragma unroll
    for (int j = 0; j < 2; ++j) {
      const float vg = vv[j] * bb[j];             // values_gated
      const float db = (1.0f - bb[j]) * mask;     // discount_beta
#pragma unroll
      for (int r = 0; r < NR; ++r) Vst[r][j] = db * Vst[r][j] + vg * cs[r];
    }

    // ---- wave32 butterfly reduce (all lanes end with full per-head sums) ----
#pragma unroll
    for (int off = 16; off > 0; off >>= 1) {
#pragma unroll
      for (int r = 0; r < NR; ++r) scoreL[r] += __shfl_xor(scoreL[r], off, 32);
      normL += __shfl_xor(normL, off, 32);
    }
    if (lane == 0) {
#pragma unroll
      for (int r = 0; r < NR; ++r) sc[h][r] = scoreL[r];
    }
    __syncthreads();

    // ---- cross-head score sum, then output ----
    float score[NR];
#pragma unroll
    for (int r = 0; r < NR; ++r) {
      float s = 0.0f;
#pragma unroll
      for (int hh = 0; hh < NHEAD; ++hh) s += sc[hh][r];
      score[r] = s;
    }
    const float inv = 1.0f / (2.0f * (float)NR * normL + EPS);
#pragma unroll
    for (int j = 0; j < 2; ++j) {
      const int d = j * 32 + lane;
      float kv = 0.0f;
#pragma unroll
      for (int r = 0; r < NR; ++r) kv += Vst[r][j] * score[r];
      attn[((size_t)t * BATCH + b) * (NHEAD * DHEAD) + h * DHEAD + d] = kv * inv;
    }
  }

  // ---- write final carried states ----
#pragma unroll
  for (int r = 0; r < NR; ++r) {
#pragma unroll
    for (int j = 0; j < 8; ++j) {
      const int f = j * 32 + lane;
      fk_out[((size_t)(b * NR + r) * NHEAD + h) * FDIM + f] = Kst[r][j];
    }
#pragma unroll
    for (int j = 0; j < 2; ++j) {
      const int d = j * 32 + lane;
      fv_out[((size_t)(b * NR + r) * NHEAD + h) * DHEAD + d] = Vst[r][j];
    }
  }
#pragma unroll
  for (int j = 0; j < 8; ++j) {
    const int f = j * 32 + lane;
    fs_out[((size_t)b * NHEAD + h) * FDIM + f] = Sst[j];
  }
  if (threadIdx.x == 0) tick_out[b] = tk + (float)T_LEN;
}

// ---------------------------------------------------------------------------
// Launch
// ---------------------------------------------------------------------------
extern "C" void kernel_launch(void* const* d_in, const int* in_sizes, int n_in,
                              void* d_out, int out_size, void* d_ws, size_t ws_size,
                              hipStream_t stream) {
  const float* inputs  = (const float*)d_in[0];   // (T,B,DM)
  const int*   term    = (const int*)  d_in[1];   // (T,B)
  const float* k_prev  = (const float*)d_in[2];   // (B,R,H,FD)
  const float* v_prev  = (const float*)d_in[3];   // (B,R,H,DH)
  const float* s_prev  = (const float*)d_in[4];   // (B,H,FD)
  const float* tick    = (const float*)d_in[5];   // (B)
  const float* Wf      = (const float*)d_in[6];   // (PROJ,DM)
  const float* bf      = (const float*)d_in[7];   // (PROJ)
  const float* Wp      = (const float*)d_in[8];   // (DM, H*DH)
  const float* bp      = (const float*)d_in[9];   // (DM)

  float* out = (float*)d_out;
  // output layout (flat, return order):
  const size_t OFF_FK   = (size_t)T_LEN * BATCH * DMODEL;              // 4,194,304
  const size_t OFF_FV   = OFF_FK + (size_t)BATCH * NR * NHEAD * FDIM;  // +262,144
  const size_t OFF_FS   = OFF_FV + (size_t)BATCH * NR * NHEAD * DHEAD; // +65,536
  const size_t OFF_TICK = OFF_FS + (size_t)BATCH * NHEAD * FDIM;       // +65,536

  // workspace: all_proj (43.5 MB) + attn_out (8 MB)
  float* proj_ws = (float*)d_ws;                                   // T*B*PROJ
  float* attn_ws = proj_ws + (size_t)T_LEN * BATCH * PROJ;         // T*B*512

  const int M = T_LEN * BATCH;   // 4096

  // K1: all_proj = inputs @ Wf^T + bf   (M=4096, N=2656, K=1024)
  wmma_gemm_nt<<<dim3(PROJ / TILE_N, M / TILE_M), 256, 0, stream>>>(
      inputs, Wf, bf, proj_ws, M, PROJ, DMODEL);

  // K2: recurrent scan
  agalite_scan<<<BATCH, 256, 0, stream>>>(
      proj_ws, term, k_prev, v_prev, s_prev, tick,
      attn_ws, out + OFF_FK, out + OFF_FV, out + OFF_FS, out + OFF_TICK);

  // K3: out = attn_out @ Wp^T + bp   (M=4096, N=1024, K=512)
  wmma_gemm_nt<<<dim3(DMODEL / TILE_N, M / TILE_M), 256, 0, stream>>>(
      attn_ws, Wp, bp, out, M, DMODEL, NHEAD * DHEAD);
}